// LinearChainCrf_43765716746609
// MI455X (gfx1250) — compile-verified
//
#include <hip/hip_runtime.h>
#include <math.h>

// CRF marginals: B=512, T=1024, K=32, SOS=30, EOS=31.
// Phase 1: 64 single-wave blocks. Blocks 0-31: forward scan (alpha -> d_out, Z1 -> ws).
//          Blocks 32-63: backward scan (betaM = beta - feats -> ws).
// Phase 2: elementwise marginal: out = mask ? exp(alpha + betaM - Z1) : 0.
//
// lse step in exp-domain with fp32 WMMA: D[i,b] = sum_k A'[i,k] * V[b,pi(k)]
// where A'[i,k] = expTrans[i, pi(k)] is constant and pi is chosen so that the
// B-layout operand of chunk kk is exactly the register pair (v[2kk], v[2kk+1])
// of the per-lane exp values -- no cross-lane data movement per step.
// pi = [0,1,8,9, 2,3,10,11, 4,5,12,13, 6,7,14,15,
//       16,17,24,25, 18,19,26,27, 20,21,28,29, 22,23,30,31]

#define NEGV (-10000.0f)
#define LOG2E 1.4426950408889634f
#define LN2 0.6931471805599453f
constexpr int Bn = 512, Tn = 1024, Kn = 32;

typedef __attribute__((ext_vector_type(2))) float v2f;
typedef __attribute__((ext_vector_type(8))) float v8f;

__device__ __forceinline__ float lane_swap16(float x) {
#if __has_builtin(__builtin_amdgcn_permlanex16)
  int i = __builtin_amdgcn_permlanex16(__float_as_int(x), __float_as_int(x),
                                       0x76543210, 0xfedcba98, false, false);
  return __int_as_float(i);
#else
  return __shfl_xor(x, 16, 32);
#endif
}

__device__ __forceinline__ float fexp2(float x) {
#if __has_builtin(__builtin_amdgcn_exp2f)
  return __builtin_amdgcn_exp2f(x);
#else
  return exp2f(x);
#endif
}

__device__ __forceinline__ float flog2(float x) {
#if __has_builtin(__builtin_amdgcn_logf)
  return __builtin_amdgcn_logf(x);
#else
  return log2f(x);
#endif
}

__device__ __forceinline__ v8f wmma4(v2f a, v2f b, v8f c) {
  // 8 args: (neg_a, A, neg_b, B, c_mod, C, reuse_a, reuse_b)
  return __builtin_amdgcn_wmma_f32_16x16x4_f32(false, a, false, b, (short)0, c,
                                               false, false);
}

// State register layout (C/D layout of 16x16 f32 WMMA, two state tiles):
//   a[r], r=0..7 : lanes 0-15 -> state r,    lanes 16-31 -> state 8+r
//   a[8+r]       : lanes 0-15 -> state 16+r, lanes 16-31 -> state 24+r
__device__ __forceinline__ void lse_step(const float (&a)[16],
                                         const v2f (&wA0)[8], const v2f (&wA1)[8],
                                         float (&out)[16]) {
  float m0 = fmaxf(a[0], a[1]), m1 = fmaxf(a[2], a[3]);
  float m2 = fmaxf(a[4], a[5]), m3 = fmaxf(a[6], a[7]);
  float m4 = fmaxf(a[8], a[9]), m5 = fmaxf(a[10], a[11]);
  float m6 = fmaxf(a[12], a[13]), m7 = fmaxf(a[14], a[15]);
  m0 = fmaxf(m0, m1); m2 = fmaxf(m2, m3); m4 = fmaxf(m4, m5); m6 = fmaxf(m6, m7);
  m0 = fmaxf(m0, m2); m4 = fmaxf(m4, m6);
  float m = fmaxf(m0, m4);
  m = fmaxf(m, lane_swap16(m));

  float v[16];
#pragma unroll
  for (int r = 0; r < 16; ++r) v[r] = fexp2((a[r] - m) * LOG2E);

  const v8f z0 = {0.f, 0.f, 0.f, 0.f, 0.f, 0.f, 0.f, 0.f};
  v8f d0a = z0, d0b = z0, d1a = z0, d1b = z0;
#pragma unroll
  for (int kk = 0; kk < 4; ++kk) {
    v2f bb; bb.x = v[2 * kk]; bb.y = v[2 * kk + 1];
    d0a = wmma4(wA0[kk], bb, d0a);
    d1a = wmma4(wA1[kk], bb, d1a);
  }
#pragma unroll
  for (int kk = 4; kk < 8; ++kk) {
    v2f bb; bb.x = v[2 * kk]; bb.y = v[2 * kk + 1];
    d0b = wmma4(wA0[kk], bb, d0b);
    d1b = wmma4(wA1[kk], bb, d1b);
  }
  const v8f d0 = d0a + d0b;
  const v8f d1 = d1a + d1b;
#pragma unroll
  for (int r = 0; r < 8; ++r) {
    out[r]     = fmaf(flog2(d0[r]), LN2, m);
    out[8 + r] = fmaf(flog2(d1[r]), LN2, m);
  }
}

__global__ __launch_bounds__(32)
void crf_scan(const float* __restrict__ feats, const float* __restrict__ masks,
              const float* __restrict__ trans, float* __restrict__ alpha,
              float* __restrict__ betaM, float* __restrict__ Z1) {
  const int lane = threadIdx.x;
  const int row  = lane & 15;
  const int hi   = lane >> 4;
  const bool lo  = (hi == 0);
  const bool isFwd = (blockIdx.x < (Bn / 16));
  const int tile = isFwd ? blockIdx.x : (blockIdx.x - Bn / 16);
  const int b = tile * 16 + row;

  // sequence length (sum of left-aligned mask)
  float part = 0.f;
  const float* mrow = masks + (size_t)b * Tn + hi * (Tn / 2);
  for (int tt = 0; tt < Tn / 2; ++tt) part += mrow[tt];
  const int nv = (int)(part + lane_swap16(part) + 0.5f);

  // Constant A tiles (A-layout, columns permuted by pi):
  // fwd: A[i,j] = exp(trans[i,j]);  bwd: A[i,j] = exp(trans[j,i])
  v2f wA0[8], wA1[8];
#pragma unroll
  for (int kk = 0; kk < 8; ++kk) {
    // pi(4kk + 2*hi) : consecutive pair of permuted columns for this operand reg
    const int jc = ((kk >> 2) << 4) + ((kk & 3) << 1) + (hi << 3);
    if (isFwd) {
      wA0[kk].x = fexp2(trans[row * Kn + jc] * LOG2E);
      wA0[kk].y = fexp2(trans[row * Kn + jc + 1] * LOG2E);
      wA1[kk].x = fexp2(trans[(16 + row) * Kn + jc] * LOG2E);
      wA1[kk].y = fexp2(trans[(16 + row) * Kn + jc + 1] * LOG2E);
    } else {
      wA0[kk].x = fexp2(trans[jc * Kn + row] * LOG2E);
      wA0[kk].y = fexp2(trans[(jc + 1) * Kn + row] * LOG2E);
      wA1[kk].x = fexp2(trans[jc * Kn + 16 + row] * LOG2E);
      wA1[kk].y = fexp2(trans[(jc + 1) * Kn + 16 + row] * LOG2E);
    }
  }
  // trans[EOS, state(r)] in the state register layout
  float tE[16];
#pragma unroll
  for (int r = 0; r < 16; ++r) {
    const int s = ((r >> 3) << 4) + (r & 7) + hi * 8;
    tE[r] = trans[(Kn - 1) * Kn + s];
  }

  const size_t lbase = (size_t)b * Tn * Kn + hi * 8;
  const float* fb = feats + lbase;

  if (isFwd) {
    float a[16];
#pragma unroll
    for (int r = 0; r < 16; ++r) {
      const int s = ((r >> 3) << 4) + (r & 7) + hi * 8;
      a[r] = (s == Kn - 2) ? 0.f : NEGV;  // onehot SOS
    }
    float* ab = alpha + lbase;
    for (int t = 0; t < Tn; ++t) {
      const float* ep = fb + (size_t)t * Kn;
      float4 e0 = *(const float4*)(ep);
      float4 e1 = *(const float4*)(ep + 4);
      float4 e2 = *(const float4*)(ep + 16);
      float4 e3 = *(const float4*)(ep + 20);
      float s[16];
      lse_step(a, wA0, wA1, s);
      const float e[16] = {e0.x, e0.y, e0.z, e0.w, e1.x, e1.y, e1.z, e1.w,
                           e2.x, e2.y, e2.z, e2.w, e3.x, e3.y, e3.z, e3.w};
      const bool upd = (t < nv);
#pragma unroll
      for (int r = 0; r < 16; ++r) a[r] = upd ? (e[r] + s[r]) : a[r];
      float* op = ab + (size_t)t * Kn;
      *(float4*)(op)      = make_float4(a[0], a[1], a[2], a[3]);
      *(float4*)(op + 4)  = make_float4(a[4], a[5], a[6], a[7]);
      *(float4*)(op + 16) = make_float4(a[8], a[9], a[10], a[11]);
      *(float4*)(op + 20) = make_float4(a[12], a[13], a[14], a[15]);
    }
    // Z1[b] = lse_j( aT[j] + trans[EOS, j] )
    float w[16];
#pragma unroll
    for (int r = 0; r < 16; ++r) w[r] = a[r] + tE[r];
    float mm = w[0];
#pragma unroll
    for (int r = 1; r < 16; ++r) mm = fmaxf(mm, w[r]);
    mm = fmaxf(mm, lane_swap16(mm));
    float sum = 0.f;
#pragma unroll
    for (int r = 0; r < 16; ++r) sum += fexp2((w[r] - mm) * LOG2E);
    sum += lane_swap16(sum);
    if (lo) Z1[b] = fmaf(flog2(sum), LN2, mm);
  } else {
    // beta[t][i] = feats[t][i] + lse_j(beta[t+1][j] + trans[j,i]); store betaM = beta - feats
    float bbv[16];
#pragma unroll
    for (int r = 0; r < 16; ++r) bbv[r] = 0.f;
    float* bmb = betaM + lbase;
    for (int t = Tn - 1; t >= 0; --t) {
      const float* ep = fb + (size_t)t * Kn;
      float4 e0 = *(const float4*)(ep);
      float4 e1 = *(const float4*)(ep + 4);
      float4 e2 = *(const float4*)(ep + 16);
      float4 e3 = *(const float4*)(ep + 20);
      float s[16];
      lse_step(bbv, wA0, wA1, s);
      const float e[16] = {e0.x, e0.y, e0.z, e0.w, e1.x, e1.y, e1.z, e1.w,
                           e2.x, e2.y, e2.z, e2.w, e3.x, e3.y, e3.z, e3.w};
      const bool isInit = (t == nv - 1);
      const bool inSeq  = (t < nv);
      float bm[16];
#pragma unroll
      for (int r = 0; r < 16; ++r) bm[r] = isInit ? tE[r] : s[r];
      float* op = bmb + (size_t)t * Kn;
      *(float4*)(op)      = make_float4(bm[0], bm[1], bm[2], bm[3]);
      *(float4*)(op + 4)  = make_float4(bm[4], bm[5], bm[6], bm[7]);
      *(float4*)(op + 16) = make_float4(bm[8], bm[9], bm[10], bm[11]);
      *(float4*)(op + 20) = make_float4(bm[12], bm[13], bm[14], bm[15]);
#pragma unroll
      for (int r = 0; r < 16; ++r) bbv[r] = inSeq ? (bm[r] + e[r]) : bbv[r];
    }
  }
}

__global__ __launch_bounds__(256)
void crf_marg(float* __restrict__ out, const float* __restrict__ betaM,
              const float* __restrict__ masks, const float* __restrict__ Z1) {
  const size_t gid = (size_t)blockIdx.x * blockDim.x + threadIdx.x;  // float4 index
  const size_t bt = gid >> 3;      // b*T + t   (8 float4 per K=32 row)
  const int b = (int)(bt >> 10);   // / Tn
  float4* po = ((float4*)out) + gid;
  float4 r;
  if (masks[bt] > 0.5f) {
    const float z = Z1[b];
    const float4 al = *po;
    const float4 bm = ((const float4*)betaM)[gid];
    r.x = fexp2((al.x + bm.x - z) * LOG2E);
    r.y = fexp2((al.y + bm.y - z) * LOG2E);
    r.z = fexp2((al.z + bm.z - z) * LOG2E);
    r.w = fexp2((al.w + bm.w - z) * LOG2E);
  } else {
    r = make_float4(0.f, 0.f, 0.f, 0.f);
  }
  *po = r;
}

extern "C" void kernel_launch(void* const* d_in, const int* in_sizes, int n_in,
                              void* d_out, int out_size, void* d_ws, size_t ws_size,
                              hipStream_t stream) {
  (void)in_sizes; (void)n_in; (void)out_size; (void)ws_size;
  const float* feats = (const float*)d_in[0];
  const float* masks = (const float*)d_in[1];
  const float* trans = (const float*)d_in[2];
  float* out = (float*)d_out;                 // alpha scratch, then marginals
  float* betaM = (float*)d_ws;                // B*T*K floats
  float* Z1 = betaM + (size_t)Bn * Tn * Kn;   // B floats

  crf_scan<<<2 * (Bn / 16), 32, 0, stream>>>(feats, masks, trans, out, betaM, Z1);
  crf_marg<<<(Bn * Tn * Kn / 4) / 256, 256, 0, stream>>>(out, betaM, masks, Z1);
}